// LSTMGCN_56238301774267
// MI455X (gfx1250) — compile-verified
//
#include <hip/hip_runtime.h>
#include <math.h>

#define FDIM 128           // F_IN == F_OUT == 128
#define GEMM_BLOCK 256     // 8 wave32 waves, one 16x16 tile each -> covers N=128

typedef float v2f __attribute__((ext_vector_type(2)));
typedef float v8f __attribute__((ext_vector_type(8)));

// ---------------------------------------------------------------------------
// WMMA helpers: D = A(16xK) * B(Kx16) + C  using V_WMMA_F32_16X16X4_F32.
// Per-lane layout (CDNA5 ISA 7.12.2, wave32):
//   A 16x4 : lane L holds row M=L%16; VGPR0/1 = K = 2*(L/16), 2*(L/16)+1
//   B 4x16 : lane L holds col N=L%16; VGPR0/1 = K = 2*(L/16), 2*(L/16)+1
//   C/D    : 8 VGPRs, element v -> M = v + 8*(L/16), N = L%16
// ---------------------------------------------------------------------------
__device__ __forceinline__ v8f wmma_accum_k(const float* __restrict__ A, int lda,
                                            const float* __restrict__ B, int ldb,
                                            int m_base, int n_base, int K, v8f acc) {
    const int lane = threadIdx.x & 31;
    const int half = lane >> 4;          // 0: lanes 0-15, 1: lanes 16-31
    const int l16  = lane & 15;
    const float* __restrict__ arow = A + (size_t)(m_base + l16) * lda;
    const float* __restrict__ bcol = B + (n_base + l16);
#pragma unroll 8
    for (int k0 = 0; k0 < K; k0 += 4) {
        const int ka = k0 + (half << 1);
        v2f a;
        a.x = arow[ka];
        a.y = arow[ka + 1];
        v2f b;
        b.x = bcol[(size_t)ka * ldb];
        b.y = bcol[(size_t)(ka + 1) * ldb];
        acc = __builtin_amdgcn_wmma_f32_16x16x4_f32(false, a, false, b,
                                                    (short)0, acc, false, false);
    }
    return acc;
}

__device__ __forceinline__ void wmma_store_tile(float* __restrict__ C, int ldc,
                                                int m_base, int n_base, v8f acc) {
    const int lane = threadIdx.x & 31;
    const int half = lane >> 4;
    const int l16  = lane & 15;
    float* __restrict__ cp = C + (size_t)(m_base + half * 8) * ldc + n_base + l16;
#pragma unroll
    for (int v = 0; v < 8; ++v) cp[(size_t)v * ldc] = acc[v];
}

// ---------------------------------------------------------------------------
// 1) degree init / accumulate / rsqrt (in place)
// ---------------------------------------------------------------------------
__global__ void lgcn_deg_init(float* __restrict__ deg, int N) {
    int i = blockIdx.x * blockDim.x + threadIdx.x;
    if (i < N) deg[i] = 1.0f;                       // self-loop weight
}

__global__ void lgcn_deg_accum(const int* __restrict__ dst,
                               const float* __restrict__ w,
                               float* __restrict__ deg, int E) {
    int e = blockIdx.x * blockDim.x + threadIdx.x;
    if (e < E) unsafeAtomicAdd(&deg[dst[e]], w[e]); // global_atomic_add_f32
}

__global__ void lgcn_dinv(float* __restrict__ deg, int N) {
    int i = blockIdx.x * blockDim.x + threadIdx.x;
    if (i < N) {
        float d = deg[i];
        deg[i] = (d > 0.0f) ? rsqrtf(fmaxf(d, 1e-12f)) : 0.0f;
    }
}

// ---------------------------------------------------------------------------
// 2) xw_g = X @ Wc_g for gates i,f,o.  grid = (M/16, 3), block = 256.
// ---------------------------------------------------------------------------
__global__ void lgcn_gemm_xw(const float* __restrict__ X,
                             const float* __restrict__ W0,
                             const float* __restrict__ W1,
                             const float* __restrict__ W2,
                             float* __restrict__ o0,
                             float* __restrict__ o1,
                             float* __restrict__ o2) {
    const int wave   = threadIdx.x >> 5;
    const int m_base = blockIdx.x * 16;
    const int n_base = wave * 16;
    const int g      = blockIdx.y;
    const float* W = (g == 0) ? W0 : (g == 1) ? W1 : W2;
    float*       O = (g == 0) ? o0 : (g == 1) ? o1 : o2;
    v8f acc = {};
    acc = wmma_accum_k(X, FDIM, W, FDIM, m_base, n_base, FDIM, acc);
    wmma_store_tile(O, FDIM, m_base, n_base, acc);
}

// ---------------------------------------------------------------------------
// 3) gconv_g = bc_g + xw_g * dinv[n]^2   (bias + self-loop message)
// ---------------------------------------------------------------------------
__global__ void lgcn_gconv_init(const float* __restrict__ xw0,
                                const float* __restrict__ xw1,
                                const float* __restrict__ xw2,
                                const float* __restrict__ bc0,
                                const float* __restrict__ bc1,
                                const float* __restrict__ bc2,
                                const float* __restrict__ dinv,
                                float* __restrict__ g0,
                                float* __restrict__ g1,
                                float* __restrict__ g2, int total) {
    int idx = blockIdx.x * blockDim.x + threadIdx.x;
    if (idx >= total) return;
    int n = idx >> 7;            // /128
    int f = idx & (FDIM - 1);
    float s = dinv[n];
    s *= s;
    g0[idx] = bc0[f] + xw0[idx] * s;
    g1[idx] = bc1[f] + xw1[idx] * s;
    g2[idx] = bc2[f] + xw2[idx] * s;
}

// ---------------------------------------------------------------------------
// 4) edge scatter: one wave32 per edge, float4 per lane, all 3 gates.
//    L2-resident gather (xw = 5 MB/gate << 192 MB L2), fp32 atomic scatter.
// ---------------------------------------------------------------------------
__global__ void lgcn_scatter(const int* __restrict__ src,
                             const int* __restrict__ dst,
                             const float* __restrict__ w,
                             const float* __restrict__ dinv,
                             const float* __restrict__ xw0,
                             const float* __restrict__ xw1,
                             const float* __restrict__ xw2,
                             float* __restrict__ g0,
                             float* __restrict__ g1,
                             float* __restrict__ g2, int E) {
    const int e = blockIdx.x * (blockDim.x >> 5) + (threadIdx.x >> 5);
    if (e >= E) return;
    const int lane = threadIdx.x & 31;
    const int s = src[e];
    const int d = dst[e];
    const float norm = dinv[s] * w[e] * dinv[d];
    const size_t soff = (size_t)s * FDIM + lane * 4;
    const size_t doff = (size_t)d * FDIM + lane * 4;

    float4 v;
    v = *(const float4*)(xw0 + soff);
    unsafeAtomicAdd(g0 + doff + 0, v.x * norm);
    unsafeAtomicAdd(g0 + doff + 1, v.y * norm);
    unsafeAtomicAdd(g0 + doff + 2, v.z * norm);
    unsafeAtomicAdd(g0 + doff + 3, v.w * norm);
    v = *(const float4*)(xw1 + soff);
    unsafeAtomicAdd(g1 + doff + 0, v.x * norm);
    unsafeAtomicAdd(g1 + doff + 1, v.y * norm);
    unsafeAtomicAdd(g1 + doff + 2, v.z * norm);
    unsafeAtomicAdd(g1 + doff + 3, v.w * norm);
    v = *(const float4*)(xw2 + soff);
    unsafeAtomicAdd(g2 + doff + 0, v.x * norm);
    unsafeAtomicAdd(g2 + doff + 1, v.y * norm);
    unsafeAtomicAdd(g2 + doff + 2, v.z * norm);
    unsafeAtomicAdd(g2 + doff + 3, v.w * norm);
}

// ---------------------------------------------------------------------------
// 5) gate pre-activations: P_g = [gconv_sel | H] @ Wl_g + bl_g
//    Wl is [256,128] row-major: rows 0-127 hit the conv part, 128-255 hit H.
//    grid = (M/16, 4), block = 256. Gate 3 (ct) reuses g_f (reference bug).
// ---------------------------------------------------------------------------
__global__ void lgcn_gemm_gate(const float* __restrict__ g0,
                               const float* __restrict__ g1,
                               const float* __restrict__ g2,
                               const float* __restrict__ H,
                               const float* __restrict__ Wl0,
                               const float* __restrict__ Wl1,
                               const float* __restrict__ Wl2,
                               const float* __restrict__ Wl3,
                               const float* __restrict__ bl0,
                               const float* __restrict__ bl1,
                               const float* __restrict__ bl2,
                               const float* __restrict__ bl3,
                               float* __restrict__ p0,
                               float* __restrict__ p1,
                               float* __restrict__ p2,
                               float* __restrict__ p3) {
    const int wave   = threadIdx.x >> 5;
    const int m_base = blockIdx.x * 16;
    const int n_base = wave * 16;
    const int gi     = blockIdx.y;
    const float* A1 = (gi == 0) ? g0 : (gi == 2) ? g2 : g1;   // i->g_i, f/ct->g_f, o->g_o
    const float* Wl = (gi == 0) ? Wl0 : (gi == 1) ? Wl1 : (gi == 2) ? Wl2 : Wl3;
    const float* bl = (gi == 0) ? bl0 : (gi == 1) ? bl1 : (gi == 2) ? bl2 : bl3;
    float*       P  = (gi == 0) ? p0  : (gi == 1) ? p1  : (gi == 2) ? p2  : p3;

    v8f acc = {};
    acc = wmma_accum_k(A1, FDIM, Wl, FDIM, m_base, n_base, FDIM, acc);
    acc = wmma_accum_k(H,  FDIM, Wl + (size_t)FDIM * FDIM, FDIM, m_base, n_base, FDIM, acc);

    const int lane = threadIdx.x & 31;
    const float b = bl[n_base + (lane & 15)];
#pragma unroll
    for (int v = 0; v < 8; ++v) acc[v] += b;
    wmma_store_tile(P, FDIM, m_base, n_base, acc);
}

// ---------------------------------------------------------------------------
// 6) LSTM cell elementwise; writes [H_new | C_new] into d_out.
// ---------------------------------------------------------------------------
__global__ void lgcn_cell(const float* __restrict__ pi,
                          const float* __restrict__ pf,
                          const float* __restrict__ po,
                          const float* __restrict__ pct,
                          const float* __restrict__ C,
                          float* __restrict__ out, int total) {
    int i = blockIdx.x * blockDim.x + threadIdx.x;
    if (i >= total) return;
    const float I  = 1.0f / (1.0f + __expf(-pi[i]));
    const float Fg = 1.0f / (1.0f + __expf(-pf[i]));
    const float O  = 1.0f / (1.0f + __expf(-po[i]));
    const float Ct = tanhf(pct[i]);
    const float Cn = Ct * I + Fg * C[i];
    const float Hn = O * tanhf(Cn);
    out[i]         = Hn;
    out[total + i] = Cn;
}

// ---------------------------------------------------------------------------
extern "C" void kernel_launch(void* const* d_in, const int* in_sizes, int n_in,
                              void* d_out, int out_size, void* d_ws, size_t ws_size,
                              hipStream_t stream) {
    const float* X    = (const float*)d_in[0];
    const int*   eidx = (const int*)d_in[1];    // [2,E] row-major
    const float* ew   = (const float*)d_in[2];
    const float* H    = (const float*)d_in[3];
    const float* C    = (const float*)d_in[4];
    const float *Wc[4], *bc[4], *Wl[4], *bl[4];
    for (int g = 0; g < 4; ++g) {
        Wc[g] = (const float*)d_in[5 + 4 * g];
        bc[g] = (const float*)d_in[6 + 4 * g];
        Wl[g] = (const float*)d_in[7 + 4 * g];
        bl[g] = (const float*)d_in[8 + 4 * g];
    }
    const int N = in_sizes[0] / FDIM;   // 10000 (multiple of 16)
    const int E = in_sizes[2];          // 640000
    const int* src = eidx;
    const int* dst = eidx + E;
    const int total = N * FDIM;

    // workspace carve-up (all regions fully rewritten every call)
    float* ws   = (float*)d_ws;
    const size_t nf = (size_t)total;
    float* deg  = ws;                                   // N (holds dinv after pass 3)
    float* base = ws + (((size_t)N + 255) & ~(size_t)255);
    float* xw0 = base;           float* xw1 = xw0 + nf; float* xw2 = xw1 + nf;
    float* gc0 = xw2 + nf;       float* gc1 = gc0 + nf; float* gc2 = gc1 + nf;
    float* p0  = gc2 + nf;       float* p1  = p0 + nf;
    float* p2  = p1 + nf;        float* p3  = p2 + nf;

    const int tilesM = N / 16;

    // 1) normalization coefficients
    lgcn_deg_init<<<(N + 255) / 256, 256, 0, stream>>>(deg, N);
    lgcn_deg_accum<<<(E + 255) / 256, 256, 0, stream>>>(dst, ew, deg, E);
    lgcn_dinv<<<(N + 255) / 256, 256, 0, stream>>>(deg, N);

    // 2) xw_g = X @ Wc_g  (WMMA fp32)
    lgcn_gemm_xw<<<dim3(tilesM, 3), GEMM_BLOCK, 0, stream>>>(
        X, Wc[0], Wc[1], Wc[2], xw0, xw1, xw2);

    // 3) bias + self-loop init of the conv accumulators
    lgcn_gconv_init<<<(total + 255) / 256, 256, 0, stream>>>(
        xw0, xw1, xw2, bc[0], bc[1], bc[2], deg, gc0, gc1, gc2, total);

    // 4) edge gather/scale/scatter-add (one wave per edge)
    lgcn_scatter<<<(E + 7) / 8, 256, 0, stream>>>(
        src, dst, ew, deg, xw0, xw1, xw2, gc0, gc1, gc2, E);

    // 5) gate GEMMs  (WMMA fp32, K = 256 split over conv part and H part)
    lgcn_gemm_gate<<<dim3(tilesM, 4), GEMM_BLOCK, 0, stream>>>(
        gc0, gc1, gc2, H,
        Wl[0], Wl[1], Wl[2], Wl[3],
        bl[0], bl[1], bl[2], bl[3],
        p0, p1, p2, p3);

    // 6) LSTM cell -> d_out = [H_new | C_new]
    lgcn_cell<<<(total + 255) / 256, 256, 0, stream>>>(
        p0, p1, p2, p3, C, (float*)d_out, total);
}